// SPDLinear_74414603371046
// MI455X (gfx1250) — compile-verified
//
#include <hip/hip_runtime.h>
#include <hip/hip_bf16.h>
#include <math.h>

// ---------------------------------------------------------------------------
// SPD linear layer on MI455X (gfx1250, wave32).
// All matrix functions are matmul-based (Newton-Schulz / series / squaring),
// every 32x32x32 matmul runs on V_WMMA_F32_16X16X4_F32.
// Block = 128 threads = 4 wave32; each wave owns one 16x16 output tile.
// Matrices live in one kernel-local LDS array; matrix handles are int
// offsets (never pointers) so all accesses lower to addrspace(3) ds ops.
// Elementwise transforms of matmul results are fused into the WMMA epilogue
// (applied in accumulator registers before the tile store).
// X tiles in the recursive scan are fetched with async global->LDS loads
// overlapped with the Newton-Schulz sqrt of M.
// ---------------------------------------------------------------------------

#define NN        32
#define LDSTR     34               // even padded stride: 8B-aligned rows, no bank conflicts
#define NELEM     (NN * LDSTR)     // 1088 floats per matrix slot
#define TPB       128
#define NS_ITERS  9                // Newton-Schulz iterations for sqrt/invsqrt
#define LOG_SQRTS 3                // inverse scaling: C^(1/8)
#define LOG_TERMS 8                // Mercator series terms
#define EXP_SQ    4                // exp scaling-and-squaring: /16, ^16
#define EW_STEP   (8 * LDSTR)      // float2 tiling: one pass covers 8 rows

typedef float v2f __attribute__((ext_vector_type(2)));
typedef float v8f __attribute__((ext_vector_type(8)));

#if defined(__has_builtin)
#if __has_builtin(__builtin_amdgcn_global_load_lds)
#define HAVE_ASYNC_LDS 1
#endif
#endif

// per-thread float2 base offset: thread owns pairs (r0+8p, c..c+1), p=0..3
__device__ __forceinline__ int ew_base() {
    const int tid = threadIdx.x;
    return ((tid >> 4) * LDSTR) + ((tid & 15) << 1);
}

// ---------------- core WMMA 32x32x32 f32 matmul with fused epilogue --------
// MODE 0: C = A*B
// MODE 1: C = 1.5I - 0.5*(A*B)            (Newton-Schulz polish)
// MODE 2: C = I + a*(A*B)                 (exp Horner step)
// MODE 3: C = A*B  and  C2 += a*(A*B)     (log power series step)
// C/C2 never alias A or B at any call site, so a single trailing barrier
// suffices (fragment reads race only against the previous barrier).
template <int MODE>
__device__ __forceinline__ void mm32_t(float* lds, int C, int A, int B,
                                       int C2, float a)
{
    const int tid  = threadIdx.x;
    const int wave = tid >> 5;
    const int lane = tid & 31;
    const int half = lane >> 4;          // 0: lanes 0-15, 1: lanes 16-31
    const int l15  = lane & 15;
    const int tm   = (wave >> 1) << 4;   // tile row base (0/16)
    const int tn   = (wave & 1) << 4;    // tile col base (0/16)

    const int arow = A + (tm + l15) * LDSTR;   // A row base for this lane
    const int bcol = B + tn + l15;             // B col base for this lane

    v8f acc = {};
#pragma unroll
    for (int kb = 0; kb < 8; ++kb) {     // K = 32 in steps of 4
        const int k0 = (kb << 2) + (half << 1);
        v2f av, bv;
        av.x = lds[arow + k0];
        av.y = lds[arow + k0 + 1];
        bv.x = lds[bcol + (k0    ) * LDSTR];
        bv.y = lds[bcol + (k0 + 1) * LDSTR];
        acc = __builtin_amdgcn_wmma_f32_16x16x4_f32(false, av, false, bv,
                                                    (short)0, acc, false, false);
    }
#pragma unroll
    for (int v = 0; v < 8; ++v) {        // C/D layout: lanes16-31 -> M = v+8
        const int row = tm + v + (half << 3);
        const int col = tn + l15;
        const int o   = row * LDSTR + col;
        float val = acc[v];
        if (MODE == 1) val = ((row == col) ? 1.5f : 0.0f) - 0.5f * val;
        if (MODE == 2) val = ((row == col) ? 1.0f : 0.0f) + a * val;
        lds[C + o] = val;
        if (MODE == 3) lds[C2 + o] += a * val;
    }
    __syncthreads();
}
__device__ __forceinline__ void mm32(float* lds, int C, int A, int B) {
    mm32_t<0>(lds, C, A, B, 0, 0.0f);
}
__device__ __forceinline__ void mm32_ns(float* lds, int C, int A, int B) {
    mm32_t<1>(lds, C, A, B, 0, 0.0f);
}
__device__ __forceinline__ void mm32_idplus(float* lds, int C, int A, int B, float a) {
    mm32_t<2>(lds, C, A, B, 0, a);
}
__device__ __forceinline__ void mm32_pow_axpy(float* lds, int C, int C2,
                                              int A, int B, float a) {
    mm32_t<3>(lds, C, A, B, C2, a);
}

// ---------------- elementwise helpers (unrolled float2, end with barrier) ---
__device__ __forceinline__ void ldm_load(float* lds, int D, const float* g) {
    const int o0 = ew_base();
    const int g0 = threadIdx.x << 1;
#pragma unroll
    for (int p = 0; p < 4; ++p)
        *(v2f*)(lds + D + o0 + p * EW_STEP) = *(const v2f*)(g + g0 + p * 256);
    __syncthreads();
}
__device__ __forceinline__ void ldm_store(float* g, const float* lds, int D) {
    const int o0 = ew_base();
    const int g0 = threadIdx.x << 1;
#pragma unroll
    for (int p = 0; p < 4; ++p)
        *(v2f*)(g + g0 + p * 256) = *(const v2f*)(lds + D + o0 + p * EW_STEP);
    __syncthreads();
}
__device__ __forceinline__ void ldm_store_scaled(float* g, const float* lds,
                                                 int D, float a) {
    const int o0 = ew_base();
    const int g0 = threadIdx.x << 1;
#pragma unroll
    for (int p = 0; p < 4; ++p) {
        v2f t = *(const v2f*)(lds + D + o0 + p * EW_STEP);
        t.x *= a; t.y *= a;
        *(v2f*)(g + g0 + p * 256) = t;
    }
    __syncthreads();
}
// async global -> LDS load of one 32x32 tile; consume after ldm_async_wait().
__device__ __forceinline__ void ldm_load_async(float* lds, int D, const float* g) {
#if defined(HAVE_ASYNC_LDS)
    const int tid = threadIdx.x;
    const int o0 = (tid >> 5) * LDSTR + (tid & 31);   // scalar (b32) mapping
#pragma unroll
    for (int it = 0; it < 8; ++it)
        __builtin_amdgcn_global_load_lds(
            (const __attribute__((address_space(1))) void*)(g + tid + it * 128),
            (__attribute__((address_space(3))) void*)(lds + D + o0 + it * 4 * LDSTR),
            4, 0, 0);
#else
    ldm_load(lds, D, g);
#endif
}
__device__ __forceinline__ void ldm_async_wait() {
#if defined(HAVE_ASYNC_LDS)
    asm volatile("s_wait_asynccnt 0" ::: "memory");
    __syncthreads();
#endif
}
__device__ __forceinline__ void mcopy(float* lds, int D, int S) {
    const int o0 = ew_base();
#pragma unroll
    for (int p = 0; p < 4; ++p) {
        const int o = o0 + p * EW_STEP;
        *(v2f*)(lds + D + o) = *(const v2f*)(lds + S + o);
    }
    __syncthreads();
}
__device__ __forceinline__ void mscale_copy(float* lds, int D, int S, float a) {
    const int o0 = ew_base();
#pragma unroll
    for (int p = 0; p < 4; ++p) {
        const int o = o0 + p * EW_STEP;
        v2f t = *(const v2f*)(lds + S + o);
        t.x *= a; t.y *= a;
        *(v2f*)(lds + D + o) = t;
    }
    __syncthreads();
}
__device__ __forceinline__ void mident(float* lds, int D) {
    const int tid = threadIdx.x;
    const int r0 = tid >> 4, c = (tid & 15) << 1;
    const int o0 = ew_base();
#pragma unroll
    for (int p = 0; p < 4; ++p) {
        const int r = r0 + p * 8;
        v2f t;
        t.x = (r == c    ) ? 1.0f : 0.0f;
        t.y = (r == c + 1) ? 1.0f : 0.0f;
        *(v2f*)(lds + D + o0 + p * EW_STEP) = t;
    }
    __syncthreads();
}
// D = I + a*S   (D != S required)
__device__ __forceinline__ void mid_plus(float* lds, int D, int S, float a) {
    const int tid = threadIdx.x;
    const int r0 = tid >> 4, c = (tid & 15) << 1;
    const int o0 = ew_base();
#pragma unroll
    for (int p = 0; p < 4; ++p) {
        const int r = r0 + p * 8;
        const int o = o0 + p * EW_STEP;
        v2f s = *(const v2f*)(lds + S + o);
        v2f t;
        t.x = ((r == c    ) ? 1.0f : 0.0f) + a * s.x;
        t.y = ((r == c + 1) ? 1.0f : 0.0f) + a * s.y;
        *(v2f*)(lds + D + o) = t;
    }
    __syncthreads();
}
__device__ __forceinline__ void msub_ident(float* lds, int D) {
    const int tid = threadIdx.x;
    if (tid < NN) lds[D + tid * LDSTR + tid] -= 1.0f;
    __syncthreads();
}
__device__ __forceinline__ void msym(float* lds, int D) {
    const int tid = threadIdx.x;
    const int r0 = tid >> 5, c = tid & 31;
    const int od0 = D + r0 * LDSTR + c;      // (r, c)
    const int ot0 = D + c * LDSTR + r0;      // (c, r)
#pragma unroll
    for (int it = 0; it < 8; ++it) {
        const int r = r0 + it * 4;
        if (c > r) {
            const int od = od0 + it * 4 * LDSTR;
            const int ot = ot0 + it * 4;
            float a = lds[od], b = lds[ot];
            float m = 0.5f * (a + b);
            lds[od] = m;
            lds[ot] = m;
        }
    }
    __syncthreads();
}
__device__ __forceinline__ float mtrace(float* lds, int A, int red) {
    const int tid = threadIdx.x;
    if (tid < 32) lds[red + tid] = lds[A + tid * LDSTR + tid];
    __syncthreads();
    float s = 0.0f;
#pragma unroll
    for (int i = 0; i < 32; ++i) s += lds[red + i];
    __syncthreads();
    return s;
}

// ---------------- Newton-Schulz coupled sqrt / invsqrt ---------------------
// A SPD. Y0 = A/tr(A) (spectrum in (0,1) => convergent), Z0 = I.
// As = sqrt(tr)*Y_k, Ais = Z_k/sqrt(tr).   As may alias A.
__device__ __forceinline__ void sqrt_invsqrt(float* lds, int A, int As, int Ais,
                                             int y, int z, int t, int u, int v,
                                             int red)
{
    float s = mtrace(lds, A, red);
    mscale_copy(lds, y, A, 1.0f / s);
    mident(lds, z);
    for (int it = 0; it < NS_ITERS; ++it) {
        mm32_ns(lds, t, z, y);       // t = (3I - Z*Y)/2   (fused polish)
        mm32(lds, u, y, t);          // Y' = Y*t
        mm32(lds, v, t, z);          // Z' = t*Z
        int tmp;
        tmp = y; y = u; u = tmp;
        tmp = z; z = v; v = tmp;
    }
    float sq = sqrtf(s);
    mscale_copy(lds, As,  y, sq);
    mscale_copy(lds, Ais, z, 1.0f / sq);
}

// ---------------- log(C) via inverse scaling + Mercator series -------------
// Destroys C. Produces S = log(C^(1/2^LOG_SQRTS))  (caller multiplies by 2^k).
__device__ __forceinline__ void mat_log(float* lds, int C, int S,
                                        int y, int z, int t, int u, int v,
                                        int dump, int red)
{
    for (int k = 0; k < LOG_SQRTS; ++k)
        sqrt_invsqrt(lds, C, /*As=*/C, /*Ais=*/dump, y, z, t, u, v, red);
    msub_ident(lds, C);              // E = C - I, ||E|| small now
    mcopy(lds, S, C);                // S = E              (j = 1)
    mcopy(lds, y, C);                // P = E
    for (int j = 2; j <= LOG_TERMS; ++j) {
        const float coef = ((j & 1) ? 1.0f : -1.0f) / (float)j;
        mm32_pow_axpy(lds, z, S, y, C, coef);   // z = P*E ; S += coef*(P*E)
        int tmp = y; y = z; z = tmp;
    }
}

// ---------------- exp(prescale*S) via scaling-and-squaring Taylor-Horner ---
// Returns the buffer offset holding the result (E or u, by swap parity).
__device__ __forceinline__ int mat_exp(float* lds, int S, int E, int xs, int u,
                                       float prescale)
{
    mscale_copy(lds, xs, S, prescale);           // X = prescale*S (|.| small)
    mid_plus(lds, E, xs, 1.0f / 8.0f);           // E = I + X/8
    for (int j = 7; j >= 1; --j) {
        mm32_idplus(lds, u, xs, E, 1.0f / (float)j);  // u = I + (X*E)/j
        int tmp = E; E = u; u = tmp;
    }
    for (int q = 0; q < EXP_SQ; ++q) {
        mm32(lds, u, E, E);
        int tmp = E; E = u; u = tmp;
    }
    return E;
}

// ---------------- geodesic tail (everything after sqrt/invsqrt of M) -------
// M <- As (Ais X Ais)^t As ; consumes X (also reused as scratch for log C).
__device__ __forceinline__ void geodesic_tail(float* lds, int M, int X, float tpar,
                                              int As, int Ais,
                                              int w0, int w1, int w2,
                                              int w3, int w4, int w5,
                                              int red)
{
    mm32(lds, w0, Ais, X);
    mm32(lds, w5, w0, Ais);                      // C = Ais X Ais
    msym(lds, w5);
    mat_log(lds, w5, /*S=*/X, w0, w1, w2, w3, w4, /*dump=*/Ais, red);
    // Ct = exp(t * 2^LOG_SQRTS * logC / 2^EXP_SQ) ^ (2^EXP_SQ)
    const float prescale = (float)(1 << LOG_SQRTS) * tpar / 16.0f;
    const int Ct = mat_exp(lds, X, /*E=*/w5, /*xs=*/w0, /*u=*/w1, prescale);
    mm32(lds, w2, As, Ct);
    mm32(lds, M, w2, As);                        // M = As Ct As
    msym(lds, M);
}

// ===========================================================================
// Kernels
// ===========================================================================

// weights: clamp, per-row cumsum ratios (fwd + reversed), row-normalized
__global__ void spd_prep_weights(const float* __restrict__ W,
                                 float* __restrict__ Wn_norm,
                                 float* __restrict__ t_fwd,
                                 float* __restrict__ t_rev)
{
    int o = threadIdx.x;
    if (o >= NN) return;
    float w[NN];
    float sum = 0.0f;
#pragma unroll
    for (int i = 0; i < NN; ++i) {
        float v = W[o * NN + i];
        v = fminf(fmaxf(v, 0.001f), 0.999f);
        w[i] = v;
        sum += v;
    }
    float c = 0.0f;
#pragma unroll
    for (int i = 0; i < NN; ++i) {
        c += w[i];
        t_fwd[o * NN + i] = w[i] / c;
        Wn_norm[o * NN + i] = w[i] / sum;
    }
    c = 0.0f;
#pragma unroll
    for (int i = 0; i < NN; ++i) {
        float wr = w[NN - 1 - i];
        c += wr;
        t_rev[o * NN + i] = wr / c;
    }
}

// L[b,i] = logm(x[b,i])   (one block per (b,i))
__global__ void __launch_bounds__(TPB)
spd_log_kernel(const float* __restrict__ x, float* __restrict__ L)
{
    __shared__ float smem[8 * NELEM + 32];
    float* lds = smem;
    const int red = 8 * NELEM;
    const size_t bi = blockIdx.x;

    const int C = 0, S = NELEM;
    ldm_load(lds, C, x + bi * (NN * NN));
    msym(lds, C);
    mat_log(lds, C, S, 2 * NELEM, 3 * NELEM, 4 * NELEM, 5 * NELEM, 6 * NELEM,
            7 * NELEM, red);
    ldm_store_scaled(L + bi * (NN * NN), lds, S, (float)(1 << LOG_SQRTS));
}

// out3[b,o,:] = sum_i Wn[o,i] * L[b,i,:]   -- a real GEMM, done with WMMA.
// grid = (B, 8): each block does one batch b and a 128-column chunk.
#define WSTR 130
__global__ void __launch_bounds__(TPB)
spd_wsum_kernel(const float* __restrict__ L, const float* __restrict__ Wn,
                float* __restrict__ out3)
{
    __shared__ float smem[NELEM + NN * WSTR];
    float* lds = smem;
    const int sW = 0, sL = NELEM;

    const int tid = threadIdx.x;
    const int b = blockIdx.x;
    const int chunk = blockIdx.y;            // 0..7, 128 columns each
    for (int i = tid; i < NN * NN; i += TPB)
        lds[sW + (i >> 5) * LDSTR + (i & 31)] = Wn[i];
    const float* Lb = L + ((size_t)b * NN) * (NN * NN) + chunk * 128;
#pragma unroll
    for (int p = 0; p < 8; ++p) {            // 16 rows x 128 cols, float2
        int i2 = tid + p * TPB;              // float2 index, 1024 total
        int r = i2 >> 6, c = (i2 & 63) << 1;
        *(v2f*)(lds + sL + r * WSTR + c) =
            *(const v2f*)(Lb + (size_t)r * (NN * NN) + c);
    }
    __syncthreads();

    const int wave = tid >> 5, lane = tid & 31;
    const int half = lane >> 4, l15 = lane & 15;
    for (int q = 0; q < 4; ++q) {
        const int tI = wave + q * 4;         // 16 tiles: 2 row x 8 col
        const int tr = tI & 1, tc = tI >> 1;
        v8f acc = {};
#pragma unroll
        for (int kb = 0; kb < 8; ++kb) {
            const int k0 = (kb << 2) + (half << 1);
            v2f av, bv;
            av.x = lds[sW + (tr * 16 + l15) * LDSTR + k0];
            av.y = lds[sW + (tr * 16 + l15) * LDSTR + k0 + 1];
            bv.x = lds[sL + (k0    ) * WSTR + tc * 16 + l15];
            bv.y = lds[sL + (k0 + 1) * WSTR + tc * 16 + l15];
            acc = __builtin_amdgcn_wmma_f32_16x16x4_f32(false, av, false, bv,
                                                        (short)0, acc, false, false);
        }
#pragma unroll
        for (int v = 0; v < 8; ++v) {
            const int o   = tr * 16 + v + (half << 3);
            const int col = chunk * 128 + tc * 16 + l15;
            out3[((size_t)b * NN + o) * (NN * NN) + col] = acc[v];
        }
    }
}

// in-place expm on out3 tiles (one block per (b,o))
__global__ void __launch_bounds__(TPB)
spd_exp_kernel(float* __restrict__ out3)
{
    __shared__ float smem[4 * NELEM];
    float* lds = smem;
    const size_t bo = blockIdx.x;
    const int S = 0, E = NELEM;
    ldm_load(lds, S, out3 + bo * (NN * NN));
    msym(lds, S);
    const int R = mat_exp(lds, S, E, 2 * NELEM, 3 * NELEM, 1.0f / 16.0f);
    ldm_store(out3 + bo * (NN * NN), lds, R);
}

// recursive Frechet mean scan (one block per (b,o); 31 sequential geodesics)
__global__ void __launch_bounds__(TPB)
spd_recfm_kernel(const float* __restrict__ x, const float* __restrict__ ttab,
                 float* __restrict__ out, int reverse)
{
    __shared__ float smem[10 * NELEM + 32];
    float* lds = smem;
    const int red = 10 * NELEM;
    const int bo = blockIdx.x;
    const int b = bo >> 5;
    const int o = bo & 31;

    const int M = 0, X = NELEM, As = 2 * NELEM, Ais = 3 * NELEM;
    const int w0 = 4 * NELEM, w1 = 5 * NELEM, w2 = 6 * NELEM;
    const int w3 = 7 * NELEM, w4 = 8 * NELEM, w5 = 9 * NELEM;

    const int i0 = reverse ? (NN - 1) : 0;
    ldm_load(lds, M, x + ((size_t)b * NN + i0) * (NN * NN));

    for (int k = 1; k < NN; ++k) {
        const int ik = reverse ? (NN - 1 - k) : k;
        // async fetch of X; consumed only after the NS sqrt of M (~30 matmuls)
        ldm_load_async(lds, X, x + ((size_t)b * NN + ik) * (NN * NN));
        if (k + 1 < NN) {   // prefetch the next step's tile into L2
            const int in_ = reverse ? (NN - 2 - k) : (k + 1);
            const float* nx = x + ((size_t)b * NN + in_) * (NN * NN);
            __builtin_prefetch(nx + threadIdx.x * 32, 0, 1);
        }
        sqrt_invsqrt(lds, M, As, Ais, w0, w1, w2, w3, w4, red);
        ldm_async_wait();                       // X now resident in LDS
        const float tk = ttab[o * NN + k];
        geodesic_tail(lds, M, X, tk, As, Ais, w0, w1, w2, w3, w4, w5, red);
    }
    ldm_store(out + (size_t)bo * (NN * NN), lds, M);
}

// ===========================================================================
extern "C" void kernel_launch(void* const* d_in, const int* in_sizes, int n_in,
                              void* d_out, int out_size, void* d_ws, size_t ws_size,
                              hipStream_t stream)
{
    (void)n_in; (void)out_size; (void)ws_size;
    const float* x = (const float*)d_in[0];       // [B, 32, 32, 32]
    const float* W = (const float*)d_in[1];       // [32, 32]
    const int B = in_sizes[0] / (NN * NN * NN);   // 256
    const size_t mat_elems  = (size_t)NN * NN;    // 1024
    const size_t out_block  = (size_t)B * NN * mat_elems;

    float* out1 = (float*)d_out;
    float* out2 = out1 + out_block;
    float* out3 = out2 + out_block;

    float* wsf   = (float*)d_ws;
    float* Lbuf  = wsf;                               // B*32*1024 floats
    float* Wn    = Lbuf + (size_t)B * NN * mat_elems; // 1024
    float* t_fwd = Wn + NN * NN;                      // 1024
    float* t_rev = t_fwd + NN * NN;                   // 1024

    // weights prep
    spd_prep_weights<<<1, 32, 0, stream>>>(W, Wn, t_fwd, t_rev);

    // fast (log-Euclidean) Frechet mean -> out3
    spd_log_kernel <<<B * NN, TPB, 0, stream>>>(x, Lbuf);
    spd_wsum_kernel<<<dim3(B, 8), TPB, 0, stream>>>(Lbuf, Wn, out3);
    spd_exp_kernel <<<B * NN, TPB, 0, stream>>>(out3);

    // recursive Frechet means -> out1 (forward), out2 (reversed)
    spd_recfm_kernel<<<B * NN, TPB, 0, stream>>>(x, t_fwd, out1, 0);
    spd_recfm_kernel<<<B * NN, TPB, 0, stream>>>(x, t_rev, out2, 1);
}